// LocalMessagePassing_19765439496856
// MI455X (gfx1250) — compile-verified
//
#include <hip/hip_runtime.h>
#include <hip/hip_bf16.h>

typedef __attribute__((ext_vector_type(2))) float v2f;
typedef __attribute__((ext_vector_type(8))) float v8f;

#define FEAT 128
#define LDSW 136   // 128 + 8 pad: rows k and k+2 differ by 2*136=272 = 16 banks -> half-waves hit disjoint banks

// ---------------- zero workspace ----------------
__global__ void zero_ws_kernel(float* __restrict__ p, long n) {
    long i = (long)blockIdx.x * blockDim.x + threadIdx.x;
    if (i < n) p[i] = 0.0f;
}

// ---------------- edge pass: one wave32 per edge ----------------
__global__ __launch_bounds__(256) void edge_kernel(
    const float* __restrict__ x,
    const long long* __restrict__ edge_index,
    const float* __restrict__ edge_len,
    const float* __restrict__ log_scale,
    const float* __restrict__ filter_w,
    const float* __restrict__ filter_b,
    float* __restrict__ agg,
    float* __restrict__ degree,
    int n_edges)
{
    const int lane = threadIdx.x & 31;
    // edge id is wave-uniform: force it into an SGPR so edge metadata loads scalarize
    unsigned e32 = (unsigned)(blockIdx.x * 8u + (threadIdx.x >> 5));
    e32 = __builtin_amdgcn_readfirstlane(e32);
    if (e32 >= (unsigned)n_edges) return;
    const long e = (long)e32;

    const long s = edge_index[e];            // sender   (scalar load)
    const long r = edge_index[n_edges + e];  // receiver (scalar load)
    const float len = edge_len[e];

    const float scale = expf(log_scale[0]);
    const float t = fmaxf(len / (scale + 1e-6f), 0.0f);
    const float decay = expf(-t * t);
    const float gate = 1.0f / (1.0f + expf(-(len * filter_w[0] + filter_b[0])));
    float w = decay * gate;
    if (!__builtin_isfinite(w)) w = 0.0f;    // nan_to_num

    // 32 lanes x float4 = 128-wide feature row, fully coalesced (L2-resident)
    const float4 xv = *(const float4*)(x + s * FEAT + lane * 4);
    float* dst = agg + r * FEAT + lane * 4;
    atomicAdd(dst + 0, w * xv.x);
    atomicAdd(dst + 1, w * xv.y);
    atomicAdd(dst + 2, w * xv.z);
    atomicAdd(dst + 3, w * xv.w);
    if (lane == 0) atomicAdd(degree + r, 1.0f);
}

// ---------------- GEMM + residual: out = x + (agg/deg) @ W ----------------
// Block = 128 threads = 4 waves; block covers 64 rows; each wave owns a 16-row
// stripe across all 128 columns (8 WMMA accumulator tiles of 16x16 f32).
__global__ __launch_bounds__(128) void gemm_wmma_kernel(
    const float* __restrict__ agg,
    const float* __restrict__ degree,
    const float* __restrict__ x,
    const float* __restrict__ Wm,
    float* __restrict__ out,
    int n_nodes)
{
    __shared__ float lW[FEAT * LDSW];

    const int tid = threadIdx.x;
    // stage W (64 KB) into LDS with async global->LDS DMA (no VGPR round-trip).
    // LDS dest operand = low 32 bits of the flat shared address (== LDS byte offset).
    for (int i = tid; i < FEAT * (FEAT / 4); i += 128) {
        const int row = i >> 5;
        const int c4  = (i & 31) << 2;
        const float* gptr = Wm + row * FEAT + c4;
        const unsigned lds_dst = (unsigned)(unsigned long long)(void*)&lW[row * LDSW + c4];
        asm volatile("global_load_async_to_lds_b128 %0, %1, off"
                     :: "v"(lds_dst), "v"(gptr) : "memory");
    }
    asm volatile("s_wait_asynccnt 0x0" ::: "memory");
    __syncthreads();

    const int wave = tid >> 5;
    const int lane = tid & 31;
    const int lh = lane & 15;   // lane within half-wave: M index for A, N index for B
    const int lg = lane >> 4;   // half-wave group: selects K pair {0,1} vs {2,3}

    const long m0 = (long)blockIdx.x * 64 + wave * 16;
    if (m0 >= n_nodes) return;  // wave-uniform exit; EXEC stays all-1s below

    const int arow = min((int)m0 + lh, n_nodes - 1);   // clamped tail rows (stores guarded later)
    const float inv_deg = 1.0f / fmaxf(degree[arow], 1.0f);

    v8f acc[8] = {};  // 8 tiles x 16x16 f32 accumulators

    for (int k0 = 0; k0 < FEAT; k0 += 4) {
        // A fragment: lane lh holds row m0+lh, K = k0 + 2*lg + {0,1}  (float2 load)
        const float2 av = *(const float2*)(agg + (long)arow * FEAT + k0 + 2 * lg);
        v2f a;
        a.x = av.x * inv_deg;
        a.y = av.y * inv_deg;

        #pragma unroll
        for (int t = 0; t < 8; ++t) {
            const int n0 = t * 16;
            // B fragment from LDS: lane lh holds col n0+lh, K = k0 + 2*lg + {0,1}
            v2f b;
            b.x = lW[(k0 + 2 * lg)     * LDSW + n0 + lh];
            b.y = lW[(k0 + 2 * lg + 1) * LDSW + n0 + lh];
            acc[t] = __builtin_amdgcn_wmma_f32_16x16x4_f32(
                false, a, false, b, (short)0, acc[t], false, false);
        }
    }

    // write back D + residual x; C/D layout: VGPR v -> row m0 + v + 8*lg, col n0 + lh
    #pragma unroll
    for (int t = 0; t < 8; ++t) {
        const int n0 = t * 16;
        #pragma unroll
        for (int v = 0; v < 8; ++v) {
            const int row = (int)m0 + v + lg * 8;
            if (row < n_nodes) {
                const long idx = (long)row * FEAT + n0 + lh;
                out[idx] = x[idx] + acc[t][v];
            }
        }
    }
}

extern "C" void kernel_launch(void* const* d_in, const int* in_sizes, int n_in,
                              void* d_out, int out_size, void* d_ws, size_t ws_size,
                              hipStream_t stream) {
    const float*     x          = (const float*)d_in[0];
    const long long* edge_index = (const long long*)d_in[1];
    const float*     edge_len   = (const float*)d_in[2];
    const float*     W_mix      = (const float*)d_in[3];
    const float*     log_scale  = (const float*)d_in[4];
    const float*     filter_w   = (const float*)d_in[5];
    const float*     filter_b   = (const float*)d_in[6];
    float*           out        = (float*)d_out;

    const int n_nodes = in_sizes[0] / FEAT;
    const int n_edges = in_sizes[2];

    float* agg    = (float*)d_ws;                    // [n_nodes, 128]
    float* degree = agg + (long)n_nodes * FEAT;      // [n_nodes]

    // zero accumulation buffers every call (atomics below)
    const long nz = (long)n_nodes * FEAT + n_nodes;
    zero_ws_kernel<<<(int)((nz + 255) / 256), 256, 0, stream>>>(agg, nz);

    // one wave per edge, 8 waves per 256-thread block
    edge_kernel<<<(n_edges + 7) / 8, 256, 0, stream>>>(
        x, edge_index, edge_len, log_scale, filter_w, filter_b,
        agg, degree, n_edges);

    // 64 rows per block
    gemm_wmma_kernel<<<(n_nodes + 63) / 64, 128, 0, stream>>>(
        agg, degree, x, W_mix, out, n_nodes);
}